// FPQuantizer_867583394396
// MI455X (gfx1250) — compile-verified
//
#include <hip/hip_runtime.h>
#include <stdint.h>

// ---------------------------------------------------------------------------
// MX-FP8(e4m3) fake-quantize, group=128, on gfx1250 (MI455X).
// Bandwidth-bound: 512 MiB traffic -> ~22.5us floor @ 23.3 TB/s.
// One wave32 == one 128-elem group (32 lanes x float4 == 512B).
// Depth-2 async global->LDS pipeline (ASYNCcnt) hides load latency across the
// absmax reduction dependency. Both streams are single-touch and exceed the
// 192MB L2, so loads and stores both use non-temporal hints.
// ---------------------------------------------------------------------------

typedef __attribute__((ext_vector_type(4))) float f4;

#define GROUP_ELEMS   128u
#define GROUP_BYTES   512u
#define WAVES_PER_BLOCK 8
#define GROUPS_PER_WAVE 8u

// gfx1250 async copy: per-lane 16B global -> LDS, GVS addressing
// (saddr u64 base + vaddr u32 byte offset). Tracked by ASYNCcnt.
// th:TH_LOAD_NT -> input streams through the caches without dirtying L2
// residency (single-touch data).
__device__ __forceinline__ void async_group_load(unsigned lds_byte_addr,
                                                 unsigned gbyte_off,
                                                 const float* base) {
    asm volatile("global_load_async_to_lds_b128 %0, %1, %2 th:TH_LOAD_NT"
                 :
                 : "v"(lds_byte_addr), "v"(gbyte_off), "s"(base)
                 : "memory");
}

__global__ __launch_bounds__(WAVES_PER_BLOCK * 32)
void fp8_fakequant_kernel(const float* __restrict__ x,
                          float* __restrict__ out,
                          unsigned num_groups) {
    // 8 waves * 2 buffers * 512B = 8KB LDS per block
    __shared__ __align__(16) float lds[WAVES_PER_BLOCK * 2 * GROUP_ELEMS];

    const unsigned lane   = threadIdx.x & 31u;
    const unsigned wib    = threadIdx.x >> 5;                    // wave in block
    const unsigned gwave  = blockIdx.x * WAVES_PER_BLOCK + wib;  // global wave id
    const unsigned nwaves = gridDim.x * WAVES_PER_BLOCK;

    // LDS byte address of this wave's double buffer. Flat shared pointers on
    // gfx1250 are {aperture_hi, lds_offset}; truncation to 32b = LDS offset.
    const unsigned ldsWaveBase =
        (unsigned)(uintptr_t)(&lds[0]) + wib * (2u * GROUP_BYTES);
    const unsigned ldsLane = lane * 16u;

    unsigned g = gwave;
    if (g >= num_groups) return;   // wave-uniform exit

    unsigned buf = 0;
    async_group_load(ldsWaveBase + ldsLane, g * GROUP_BYTES + ldsLane, x);

    const float kInvMax   = 1.0f / 448.0f;  // MAX_NORM = 448
    const float kShift    = 0.03125f;       // 2^(mbits-2)/2^emax = 8/256 = 1/32
    const float kInvShift = 32.0f;

    for (;;) {
        const unsigned gn  = g + nwaves;
        const bool     pre = (gn < num_groups);   // wave-uniform
        if (pre) {
            // prefetch next group into the other buffer, keep 1 op in flight;
            // async loads complete in order, so waiting <=1 releases the
            // buffer for the current group.
            async_group_load(ldsWaveBase + ((buf ^ 1u) * GROUP_BYTES) + ldsLane,
                             gn * GROUP_BYTES + ldsLane, x);
            asm volatile("s_wait_asynccnt 0x1" ::: "memory");
        } else {
            asm volatile("s_wait_asynccnt 0x0" ::: "memory");
        }

        // current group now resident in LDS: ds_load_b128 per lane
        const unsigned fidx = (wib * 2u + buf) * GROUP_ELEMS + lane * 4u;
        f4 v = *(const f4*)&lds[fidx];

        // per-lane absmax of 4, then wave32 butterfly max (group absmax)
        float m = fmaxf(fmaxf(fabsf(v.x), fabsf(v.y)),
                        fmaxf(fabsf(v.z), fabsf(v.w)));
        m = fmaxf(m, __shfl_xor(m, 16, 32));
        m = fmaxf(m, __shfl_xor(m,  8, 32));
        m = fmaxf(m, __shfl_xor(m,  4, 32));
        m = fmaxf(m, __shfl_xor(m,  2, 32));
        m = fmaxf(m, __shfl_xor(m,  1, 32));

        const float scale = m * kInvMax;
        const float rs    = (m > 0.0f) ? (kShift / scale) : 0.0f; // x -> shifted q
        const float os    = scale * kInvShift;                    // q -> dequant

        // round-nearest, ties away from zero: sign(x)*floor(|x|*rs + 0.5)
        f4 q;
        q.x = copysignf(floorf(fabsf(v.x) * rs + 0.5f), v.x) * os;
        q.y = copysignf(floorf(fabsf(v.y) * rs + 0.5f), v.y) * os;
        q.z = copysignf(floorf(fabsf(v.z) * rs + 0.5f), v.z) * os;
        q.w = copysignf(floorf(fabsf(v.w) * rs + 0.5f), v.w) * os;

        // streaming store: global_store_b128 with non-temporal hint
        f4* dst = (f4*)(out + (size_t)g * GROUP_ELEMS) + lane;
        __builtin_nontemporal_store(q, dst);

        if (!pre) break;
        g = gn;
        buf ^= 1u;
    }
}

extern "C" void kernel_launch(void* const* d_in, const int* in_sizes, int n_in,
                              void* d_out, int out_size, void* d_ws, size_t ws_size,
                              hipStream_t stream) {
    const float* x   = (const float*)d_in[0];
    float*       out = (float*)d_out;

    const long long n          = (long long)in_sizes[0];     // 4*4096*4096
    const unsigned  num_groups = (unsigned)(n / GROUP_ELEMS);

    // ~8 groups per wave amortizes the async-pipeline fill; 8 waves/block.
    unsigned nwaves = (num_groups + GROUPS_PER_WAVE - 1u) / GROUPS_PER_WAVE;
    unsigned blocks = (nwaves + WAVES_PER_BLOCK - 1u) / WAVES_PER_BLOCK;
    if (blocks == 0) blocks = 1;

    fp8_fakequant_kernel<<<blocks, WAVES_PER_BLOCK * 32, 0, stream>>>(
        x, out, num_groups);
}